// Attention_90658169684614
// MI455X (gfx1250) — compile-verified
//
#include <hip/hip_runtime.h>

// Problem constants (from reference): B=64, S=2048, D_IN=512, H=1024
#define B_    64
#define S_    2048
#define DIN_  512
#define H_    1024
#define NSPLIT 16                 // S-splits per batch (one wave32 each)
#define ROWS_PER (S_ / NSPLIT)    // 128 rows per wave

typedef float f32x2 __attribute__((ext_vector_type(2)));
typedef float f32x4 __attribute__((ext_vector_type(4)));
typedef float f32x8 __attribute__((ext_vector_type(8)));

// exp(x) via v_exp_f32 (base-2)
__device__ __forceinline__ float fexp(float x) {
    return __builtin_amdgcn_exp2f(x * 1.4426950408889634f);
}
// tanh(x) = 1 - 2/(exp2(2x*log2e)+1)   (saturates correctly at +-1)
__device__ __forceinline__ float ftanh(float x) {
    float e = __builtin_amdgcn_exp2f(x * 2.8853900817779268f);
    return 1.0f - 2.0f * __builtin_amdgcn_rcpf(e + 1.0f);
}

// ---------------------------------------------------------------------------
// Kernel 1: x = inp[64,512] @ W[512,1024] + b  via V_WMMA_F32_16X16X4_F32.
// One 16x16 output tile per wave; 256 tiles -> 64 blocks x 4 waves.
// A 16x4 f32 layout: lanes 0-15 hold K={0,1} (vgpr0/1), lanes 16-31 K={2,3}.
// B 4x16 f32 layout: symmetric (N = lane%16).
// D 16x16 f32: vgpr r, lanes 0-15 -> M=r, lanes 16-31 -> M=r+8, N=lane%16.
// ---------------------------------------------------------------------------
__global__ void k_gemm_wmma(const float* __restrict__ inp,
                            const float* __restrict__ W,
                            const float* __restrict__ bias,
                            float* __restrict__ xout) {
    int wid  = blockIdx.x * 4 + (threadIdx.x >> 5);
    int lane = threadIdx.x & 31;
    int mt = wid >> 6;          // 0..3   (M = 64)
    int nt = wid & 63;          // 0..63  (N = 1024)
    int half = lane >> 4;       // 0/1 -> K pair select
    int lm   = lane & 15;

    const float* arow = inp + (size_t)(mt * 16 + lm) * DIN_;
    f32x8 acc = (f32x8)0.0f;

    #pragma unroll 4
    for (int k0 = 0; k0 < DIN_; k0 += 4) {
        int ka = k0 + half * 2;
        f32x2 a, bm;
        a[0]  = arow[ka];
        a[1]  = arow[ka + 1];
        bm[0] = W[(size_t)ka * H_ + nt * 16 + lm];
        bm[1] = W[(size_t)(ka + 1) * H_ + nt * 16 + lm];
        acc = __builtin_amdgcn_wmma_f32_16x16x4_f32(
            /*neg_a=*/false, a, /*neg_b=*/false, bm,
            /*c_mod=*/(short)0, acc, /*reuse_a=*/false, /*reuse_b=*/false);
    }

    int n  = nt * 16 + lm;
    float bv = bias[n];
    #pragma unroll
    for (int r = 0; r < 8; ++r) {
        int m = mt * 16 + half * 8 + r;
        xout[(size_t)m * H_ + n] = acc[r] + bv;
    }
}

// ---------------------------------------------------------------------------
// Kernel 2: fused scores + online-softmax + weighted context accumulation.
// One wave32 per (batch, S-split). Lane owns h = lane*32 .. lane*32+31.
// Masked rows are skipped entirely (weight is exactly 0) -> ~half the HBM
// traffic never happens. Context streamed once with non-temporal b128 loads.
// Emits per-split partials (m, l, acc[1024]) and raw scores.
// ---------------------------------------------------------------------------
__global__ void k_attn_partial(const float* __restrict__ ctx,
                               const int*   __restrict__ mask,
                               const float* __restrict__ v,
                               const float* __restrict__ x,
                               float* __restrict__ ws_m,
                               float* __restrict__ ws_l,
                               float* __restrict__ ws_acc,
                               float* __restrict__ scores_raw) {
    int lane = threadIdx.x;          // 0..31
    int b  = blockIdx.x >> 4;        // batch
    int sp = blockIdx.x & (NSPLIT - 1);
    int hbase = lane * 32;

    // Preload this lane's x and v slices (constant across rows).
    float xl[32], vl[32];
    const f32x4* xb4 = (const f32x4*)(x + (size_t)b * H_ + hbase);
    const f32x4* v4  = (const f32x4*)(v + hbase);
    #pragma unroll
    for (int j = 0; j < 8; ++j) {
        f32x4 q = xb4[j];
        f32x4 r = v4[j];
        #pragma unroll
        for (int i2 = 0; i2 < 4; ++i2) { xl[4*j+i2] = q[i2]; vl[4*j+i2] = r[i2]; }
    }

    float acc[32];
    #pragma unroll
    for (int i = 0; i < 32; ++i) acc[i] = 0.0f;
    float m = -3.0e38f, l = 0.0f;

    float* srow = scores_raw + (size_t)b * S_;
    int s0 = sp * ROWS_PER;

    for (int s = s0; s < s0 + ROWS_PER; ++s) {
        if (mask[b * S_ + s]) {              // uniform across wave
            if (lane == 0) srow[s] = -3.0e38f;
            continue;                         // weight==0: skip ctx read
        }
        const f32x4* crow = (const f32x4*)(ctx + ((size_t)(b * S_ + s)) * H_ + hbase);
        float c[32];
        float part = 0.0f;
        #pragma unroll
        for (int j = 0; j < 8; ++j) {
            f32x4 q = __builtin_nontemporal_load(crow + j);
            #pragma unroll
            for (int i2 = 0; i2 < 4; ++i2) {
                int i = 4 * j + i2;
                c[i] = q[i2];
                part += ftanh(xl[i] + c[i]) * vl[i];
            }
        }
        // wave32 butterfly reduction -> all lanes hold full score
        #pragma unroll
        for (int off = 16; off >= 1; off >>= 1)
            part += __shfl_xor(part, off, 32);
        if (lane == 0) srow[s] = part;

        // online softmax update (rescale only on new max)
        if (part > m) {
            float cr = fexp(m - part);
            l *= cr;
            #pragma unroll
            for (int i = 0; i < 32; ++i) acc[i] *= cr;
            m = part;
        }
        float w = fexp(part - m);
        l += w;
        #pragma unroll
        for (int i = 0; i < 32; ++i) acc[i] = fmaf(w, c[i], acc[i]);
    }

    int p = blockIdx.x;  // 0..1023 partial id
    if (lane == 0) { ws_m[p] = m; ws_l[p] = l; }
    f32x4* pa = (f32x4*)(ws_acc + (size_t)p * H_ + hbase);
    #pragma unroll
    for (int j = 0; j < 8; ++j) {
        f32x4 q;
        #pragma unroll
        for (int i2 = 0; i2 < 4; ++i2) q[i2] = acc[4*j+i2];
        pa[j] = q;
    }
}

// ---------------------------------------------------------------------------
// Kernel 3: merge the 16 partials per batch, normalize, write both outputs.
// ---------------------------------------------------------------------------
__global__ void k_combine(const float* __restrict__ ws_m,
                          const float* __restrict__ ws_l,
                          const float* __restrict__ ws_acc,
                          float* __restrict__ applied,
                          float* __restrict__ weights) {
    int b = blockIdx.x;
    int tid = threadIdx.x;
    __shared__ float sm[NSPLIT], sl[NSPLIT];
    if (tid < NSPLIT) {
        sm[tid] = ws_m[b * NSPLIT + tid];
        sl[tid] = ws_l[b * NSPLIT + tid];
    }
    __syncthreads();

    float m_g = -3.0e38f;
    #pragma unroll
    for (int p = 0; p < NSPLIT; ++p) m_g = fmaxf(m_g, sm[p]);
    float l_g = 0.0f;
    float coef[NSPLIT];
    #pragma unroll
    for (int p = 0; p < NSPLIT; ++p) {
        coef[p] = fexp(sm[p] - m_g);
        l_g += sl[p] * coef[p];
    }
    float inv_l = __builtin_amdgcn_rcpf(l_g);
    #pragma unroll
    for (int p = 0; p < NSPLIT; ++p) coef[p] *= inv_l;

    // attn_applied[b, h] = sum_p acc_p[h] * exp(m_p - m_g) / l_g
    for (int h = tid; h < H_; h += 256) {
        float a = 0.0f;
        #pragma unroll
        for (int p = 0; p < NSPLIT; ++p)
            a = fmaf(ws_acc[((size_t)(b * NSPLIT + p)) * H_ + h], coef[p], a);
        applied[(size_t)b * H_ + h] = a;
    }
    // attn_weights[b, s] = exp(score - m_g) / l_g  (raw score was staged here)
    for (int s = tid; s < S_; s += 256) {
        float sc = weights[(size_t)b * S_ + s];
        weights[(size_t)b * S_ + s] = fexp(sc - m_g) * inv_l;
    }
}

// ---------------------------------------------------------------------------
extern "C" void kernel_launch(void* const* d_in, const int* in_sizes, int n_in,
                              void* d_out, int out_size, void* d_ws, size_t ws_size,
                              hipStream_t stream) {
    const float* inp  = (const float*)d_in[0];  // [64,1,512]
    // d_in[1] = hidden: unused by the reference
    const float* ctx  = (const float*)d_in[2];  // [64,2048,1024]
    const int*   mask = (const int*)  d_in[3];  // [64,1,2048] bool->int
    const float* W    = (const float*)d_in[4];  // [512,1024]
    const float* bias = (const float*)d_in[5];  // [1024]
    const float* v    = (const float*)d_in[6];  // [1024,1]

    float* out     = (float*)d_out;
    float* applied = out;            // 64*1024 floats
    float* weights = out + B_ * H_;  // 64*2048 floats

    float* ws     = (float*)d_ws;
    float* ws_x   = ws;                       // 64*1024
    float* ws_m   = ws_x + B_ * H_;           // 64*16
    float* ws_l   = ws_m + B_ * NSPLIT;       // 64*16
    float* ws_acc = ws_l + B_ * NSPLIT;       // 64*16*1024

    k_gemm_wmma   <<<64,          128, 0, stream>>>(inp, W, bias, ws_x);
    k_attn_partial<<<B_ * NSPLIT,  32, 0, stream>>>(ctx, mask, v, ws_x,
                                                    ws_m, ws_l, ws_acc, weights);
    k_combine     <<<B_,          256, 0, stream>>>(ws_m, ws_l, ws_acc,
                                                    applied, weights);
}